// SpanFusionLM_64939905515676
// MI455X (gfx1250) — compile-verified
//
#include <hip/hip_runtime.h>
#include <stdint.h>

// ---------------------------------------------------------------------------
// MI455X (gfx1250) implementation.
//  kernel 1: one workgroup per (b,k) row. Row (201KB) staged into the 320KB
//            WGP LDS once via async global->LDS B128 copies (ASYNCcnt), then
//            3 vectorized (ds_load_b128) LDS passes:
//              A: max(x)                       -> mx = clip(max)
//              B: sum e, sum x*e (entropy) + exp-mass sub-histograms
//              scan: parallel 512-bin prefix scan -> top-p threshold AND
//                    kept mass S2 (cum at crossing bin) -- no extra pass
//              D: write probs (aligned float4) + Gumbel-max sample
//            Cross-wave reductions use V_WMMA_F32_16X16X4_F32 (ones trick).
//  kernel 2: per-batch top-M_t entropy pick + seq splice.
//  HBM traffic = 1x read logits + 1x write probs (~412 MB) -> ~18us roofline;
//  per-block LDS/VALU work trimmed so the stream stays the limiter.
// ---------------------------------------------------------------------------

#define BLK    1024
#define NBIN   512
#define NSUB   4
#define HRANGE 25.0f
#define TOPP   0.9f

#define AS1 __attribute__((address_space(1)))
#define AS3 __attribute__((address_space(3)))

typedef __attribute__((ext_vector_type(2))) float v2f;
typedef __attribute__((ext_vector_type(8))) float v8f;
typedef __attribute__((ext_vector_type(4))) int   v4i;

__device__ __forceinline__ float clip50(float x) { return fminf(fmaxf(x, -50.0f), 50.0f); }

__device__ __forceinline__ float u01(unsigned x) {
  x ^= x >> 16; x *= 0x7feb352du;
  x ^= x >> 15; x *= 0x846ca68bu;
  x ^= x >> 16;
  return ((float)(x >> 8) + 0.5f) * (1.0f / 16777216.0f);
}

// ---- CDNA5 async global->LDS copy (tracked by ASYNCcnt) -------------------
__device__ __forceinline__ void cp_async16(float* lds_dst, const float* gsrc) {
#if __has_builtin(__builtin_amdgcn_global_load_async_to_lds_b128)
  __builtin_amdgcn_global_load_async_to_lds_b128(
      (AS1 v4i*)(uintptr_t)gsrc,
      (AS3 v4i*)(unsigned)(uintptr_t)lds_dst,
      0, 0);
#else
  asm volatile("global_load_async_to_lds_b128 %0, %1, off"
               :: "v"((unsigned)(uintptr_t)lds_dst),
                  "v"((unsigned long long)(uintptr_t)gsrc)
               : "memory");
#endif
}

__device__ __forceinline__ void wait_async0() {
#if __has_builtin(__builtin_amdgcn_s_wait_asynccnt)
  __builtin_amdgcn_s_wait_asynccnt(0);
#else
  asm volatile("s_wait_asynccnt 0" ::: "memory");
#endif
}

// ---- sum of 64 LDS floats, executed by wave 0 (all 32 lanes active) -------
// V_WMMA_F32_16X16X4_F32 with A = ones: D[m,n] = sum_k B[k,n]; summing d[0]
// over all 32 lanes double-counts every column once -> *0.5. Layout-invariant.
__device__ __forceinline__ float wave0_sum64(const float* s, int lane) {
#if __has_builtin(__builtin_amdgcn_wmma_f32_16x16x4_f32)
  v2f a; a.x = 1.0f; a.y = 1.0f;
  v2f b; b.x = s[2 * lane]; b.y = s[2 * lane + 1];
  v8f c = {0.f, 0.f, 0.f, 0.f, 0.f, 0.f, 0.f, 0.f};
  v8f d = __builtin_amdgcn_wmma_f32_16x16x4_f32(false, a, false, b, (short)0, c, false, false);
  float t = d[0];
  #pragma unroll
  for (int xo = 16; xo > 0; xo >>= 1) t += __shfl_xor(t, xo, 32);
  return 0.5f * t;
#else
  float t = s[2 * lane] + s[2 * lane + 1];
  #pragma unroll
  for (int xo = 16; xo > 0; xo >>= 1) t += __shfl_xor(t, xo, 32);
  return t;
#endif
}

__global__ __launch_bounds__(BLK) void k_row_softmax(
    const float* __restrict__ logits, float* __restrict__ ent_out,
    float* __restrict__ probs, int* __restrict__ toks, int V)
{
  extern __shared__ float smem[];
  float* lrow  = smem;                     // V + 8 floats (shifted row image)
  float* red   = smem + V + 8;             // 64 (upper 32 stay zero for WMMA)
  float* red2  = red + 64;                 // 64
  float* hist  = red2 + 64;                // NSUB * NBIN sub-histograms
  float* bestv = hist + NSUB * NBIN;       // 32
  int*   besti = (int*)(bestv + 32);       // 32
  float* scal  = (float*)(besti + 32);     // 8 broadcast scalars
  float* wsum  = scal + 8;                 // 16 (scan wave sums)
  float* wpre  = wsum + 16;                // 16 (scan wave prefixes)
  int*   seli  = (int*)(wpre + 16);        // 1  (selected bin, atomicMin)

  const int row  = blockIdx.x;
  const int tid  = threadIdx.x;
  const int lane = tid & 31;
  const int wave = tid >> 5;

  const float* g    = logits + (size_t)row * V;
  float*       prow = probs  + (size_t)row * V;

  for (int i = tid; i < NSUB * NBIN; i += BLK) hist[i] = 0.0f;
  if (tid < 64) { red[tid] = 0.0f; red2[tid] = 0.0f; }
  if (tid == 0) seli[0] = NBIN - 1;

  // ---- alignment bookkeeping (V=50257 -> rows are not 16B aligned) -------
  const int sh4  = (int)(((uintptr_t)g >> 2) & 3); // misalignment in floats
  const int lead = (4 - sh4) & 3;                  // scalar head elements
  const int q    = sh4;                            // elem i lives at lrow[q+i]
  const int M    = (V - lead) >> 2;                // aligned 16B chunks
  const int tailStart = lead + 4 * M;
  const int tailN     = V - tailStart;

  // head / tail via plain LDS stores (disjoint from async targets)
  if (tid < lead) lrow[q + tid] = g[tid];
  if (tid >= 64 && tid < 64 + tailN) { int i = tailStart + (tid - 64); lrow[q + i] = g[i]; }

  // bulk: async 16B global->LDS copies (ASYNCcnt); chunks are 16B aligned
  for (int m = tid; m < M; m += BLK) {
    int i0 = lead + 4 * m;
    cp_async16(&lrow[q + i0], g + i0);
  }
  wait_async0();
  __syncthreads();

  // ---- pass A: raw max (clip hoisted out: max commutes with clamp) -------
  float mraw = -1.0e30f;
  if (tid < lead) mraw = fmaxf(mraw, lrow[q + tid]);
  if (tid >= 64 && tid < 64 + tailN) mraw = fmaxf(mraw, lrow[q + tailStart + (tid - 64)]);
  for (int m = tid; m < M; m += BLK) {
    float4 v = *(const float4*)&lrow[q + lead + 4 * m];  // ds_load_b128
    mraw = fmaxf(mraw, fmaxf(fmaxf(v.x, v.y), fmaxf(v.z, v.w)));
  }
  #pragma unroll
  for (int xo = 16; xo > 0; xo >>= 1) mraw = fmaxf(mraw, __shfl_xor(mraw, xo, 32));
  if (lane == 0) red[wave] = mraw;
  __syncthreads();
  if (tid < 32) {
    float m = red[lane];
    #pragma unroll
    for (int xo = 16; xo > 0; xo >>= 1) m = fmaxf(m, __shfl_xor(m, xo, 32));
    if (lane == 0) scal[0] = clip50(m);
  }
  __syncthreads();
  const float mx = scal[0];

  // ---- pass B: entropy sums + exp-mass sub-histograms --------------------
  const float binScale = (float)NBIN / HRANGE;
  float* myhist = hist + (lane & (NSUB - 1)) * NBIN;  // cut same-addr conflicts
  float se = 0.0f, sxe = 0.0f;

  if (tid < lead) {
    float xc = clip50(lrow[q + tid]);
    float e = __expf(xc - mx);
    se += e; sxe += xc * e;
    int bin = (int)((mx - xc) * binScale); bin = bin < NBIN - 1 ? bin : NBIN - 1;
    atomicAdd(&myhist[bin], e);
  }
  if (tid >= 64 && tid < 64 + tailN) {
    float xc = clip50(lrow[q + tailStart + (tid - 64)]);
    float e = __expf(xc - mx);
    se += e; sxe += xc * e;
    int bin = (int)((mx - xc) * binScale); bin = bin < NBIN - 1 ? bin : NBIN - 1;
    atomicAdd(&myhist[bin], e);
  }
  for (int m = tid; m < M; m += BLK) {
    float4 v = *(const float4*)&lrow[q + lead + 4 * m];  // ds_load_b128
    #pragma unroll
    for (int j = 0; j < 4; ++j) {
      float xc = clip50((&v.x)[j]);
      float e = __expf(xc - mx);
      se += e; sxe += xc * e;
      int bin = (int)((mx - xc) * binScale); bin = bin < NBIN - 1 ? bin : NBIN - 1;
      atomicAdd(&myhist[bin], e);                        // ds_add_f32
    }
  }
  #pragma unroll
  for (int xo = 16; xo > 0; xo >>= 1) { se += __shfl_xor(se, xo, 32); sxe += __shfl_xor(sxe, xo, 32); }
  if (lane == 0) { red[wave] = se; red2[wave] = sxe; }
  __syncthreads();
  if (tid < 32) {
    float Se  = wave0_sum64(red,  lane);   // v_wmma_f32_16x16x4_f32
    float Sxe = wave0_sum64(red2, lane);   // v_wmma_f32_16x16x4_f32
    if (lane == 0) {
      scal[1] = Se;
      float ent = logf(Se) + mx - Sxe / Se;  // logsumexp - E[x]
      ent_out[row] = fminf(fmaxf(ent, 0.0f), 20.0f);
    }
  }
  __syncthreads();
  const float Se = scal[1];

  // ---- parallel histogram scan: threshold bin AND kept mass S2 -----------
  float cumv = 0.0f;
  if (tid < NBIN) {
    float h = 0.0f;
    #pragma unroll
    for (int s = 0; s < NSUB; ++s) h += hist[s * NBIN + tid];
    cumv = h;
    #pragma unroll
    for (int xo = 1; xo < 32; xo <<= 1) {           // wave inclusive scan
      float o = __shfl_up(cumv, xo, 32);
      if (lane >= xo) cumv += o;
    }
    if (lane == 31) wsum[tid >> 5] = cumv;          // 16 wave totals
  }
  __syncthreads();
  if (tid < 16) {                                   // tiny serial wave-prefix
    float p = 0.0f;
    for (int j = 0; j < tid; ++j) p += wsum[j];
    wpre[tid] = p;
  }
  __syncthreads();
  if (tid < NBIN) {
    cumv += wpre[tid >> 5];                         // full inclusive cum
    hist[tid] = cumv;                               // stash cumulative
    if (cumv > TOPP * Se) atomicMin(seli, tid);     // first crossing bin
  }
  __syncthreads();
  if (tid == 0) {
    int bsel = seli[0];
    scal[2] = mx - (float)(bsel + 1) * (HRANGE / (float)NBIN); // keep x >= thr
    scal[3] = 1.0f / hist[bsel];                    // S2 = kept mass (exact)
  }
  __syncthreads();
  const float thr   = scal[2];
  const float invS2 = scal[3];

  // ---- pass D: write probs (aligned float4) + Gumbel-max sample ----------
  float bv = -1.0e30f; int bi = 0;
  const unsigned seed0 = (unsigned)row * 0x9E3779B9u;

  if (tid < lead) {
    int i = tid;
    float xc = clip50(lrow[q + i]);
    float p = 0.0f;
    if (xc >= thr) {
      p = __expf(xc - mx) * invS2;
      float sc = xc + (-logf(-logf(u01(seed0 + (unsigned)i))));
      if (sc > bv) { bv = sc; bi = i; }
    }
    prow[i] = p;
  }
  if (tid >= 64 && tid < 64 + tailN) {
    int i = tailStart + (tid - 64);
    float xc = clip50(lrow[q + i]);
    float p = 0.0f;
    if (xc >= thr) {
      p = __expf(xc - mx) * invS2;
      float sc = xc + (-logf(-logf(u01(seed0 + (unsigned)i))));
      if (sc > bv) { bv = sc; bi = i; }
    }
    prow[i] = p;
  }
  for (int m = tid; m < M; m += BLK) {
    int i0 = lead + 4 * m;
    float4 v = *(const float4*)&lrow[q + i0];        // ds_load_b128
    float4 o4;
    #pragma unroll
    for (int j = 0; j < 4; ++j) {
      int i = i0 + j;
      float xc = clip50((&v.x)[j]);
      float p = 0.0f;
      if (xc >= thr) {
        p = __expf(xc - mx) * invS2;
        float sc = xc + (-logf(-logf(u01(seed0 + (unsigned)i))));
        if (sc > bv) { bv = sc; bi = i; }
      }
      (&o4.x)[j] = p;
    }
    *(float4*)(prow + i0) = o4;                      // aligned 16B store
  }

  #pragma unroll
  for (int xo = 16; xo > 0; xo >>= 1) {
    float ov = __shfl_xor(bv, xo, 32);
    int   oi = __shfl_xor(bi, xo, 32);
    if (ov > bv || (ov == bv && oi < bi)) { bv = ov; bi = oi; }
  }
  if (lane == 0) { bestv[wave] = bv; besti[wave] = bi; }
  __syncthreads();
  if (tid == 0) {
    float bb = bestv[0]; int ii = besti[0];
    for (int w = 1; w < BLK / 32; ++w) {
      if (bestv[w] > bb || (bestv[w] == bb && besti[w] < ii)) { bb = bestv[w]; ii = besti[w]; }
    }
    toks[row] = ii;
  }
}

// ---------------------------------------------------------------------------
// kernel 2: rank-based top-M_t entropy pick (stable tie-break = argsort) and
// sequence splice. 16 blocks x 64 threads — negligible cost.
// ---------------------------------------------------------------------------
__global__ void k_pick_seq(const float* __restrict__ ent, const int* __restrict__ filled,
                           const long long* __restrict__ M_t, const long long* __restrict__ seq,
                           const int* __restrict__ toks, float* __restrict__ out_seq,
                           float* __restrict__ out_pick, int K, int NP, int T)
{
  __shared__ float es[128];
  __shared__ int   avs[128];
  const int b = blockIdx.x, k = threadIdx.x;

  const int  av = (filled[b * K + k] == 0) ? 1 : 0;
  const float e = ent[b * K + k];
  es[k]  = e;
  avs[k] = av;
  __syncthreads();

  int rank = 0, nav = 0;
  for (int j = 0; j < K; ++j) {
    if (avs[j]) {
      ++nav;
      float ej = es[j];
      if (ej < e || (ej == e && j < k)) ++rank;
    }
  }
  long long m = M_t[b];
  if (m > (long long)nav) m = (long long)nav;
  const int pick = (av && rank < (int)m) ? 1 : 0;
  out_pick[b * K + k] = (float)pick;

  for (int i = k; i < NP; i += K)
    out_seq[(size_t)b * T + i] = (float)seq[(size_t)b * T + i];
  long long old = seq[(size_t)b * T + NP + k];
  out_seq[(size_t)b * T + NP + k] = pick ? (float)toks[b * K + k] : (float)old;
}

// ---------------------------------------------------------------------------
extern "C" void kernel_launch(void* const* d_in, const int* in_sizes, int n_in,
                              void* d_out, int out_size, void* d_ws, size_t ws_size,
                              hipStream_t stream) {
  (void)n_in; (void)out_size; (void)ws_size;
  const float*     logits = (const float*)d_in[0];
  const int*       filled = (const int*)d_in[1];
  const long long* M_t    = (const long long*)d_in[2];
  const long long* seq    = (const long long*)d_in[3];
  // d_in[4] = n_prompt scalar (device); NP derived from shapes instead.

  const int B  = in_sizes[2];
  const int BK = in_sizes[1];
  const int K  = BK / B;
  const int V  = in_sizes[0] / BK;
  const int T  = in_sizes[3] / B;
  const int NP = T - K;

  float* out_seq   = (float*)d_out;                 // B*T
  float* out_pick  = out_seq + (size_t)B * T;       // B*K
  float* out_ent   = out_pick + BK;                 // B*K
  float* out_probs = out_ent + BK;                  // B*K*V
  int*   toks      = (int*)d_ws;                    // B*K ints

  const size_t shbytes =
      (size_t)(V + 8 + 64 + 64 + NSUB * NBIN + 32 + 32 + 8 + 16 + 16 + 4) *
      sizeof(float);

  hipError_t err = hipFuncSetAttribute(
      reinterpret_cast<const void*>(k_row_softmax),
      hipFuncAttributeMaxDynamicSharedMemorySize, (int)shbytes);
  (void)err;

  k_row_softmax<<<dim3(BK), dim3(BLK), shbytes, stream>>>(
      logits, out_ent, out_probs, toks, V);
  k_pick_seq<<<dim3(B), dim3(K), 0, stream>>>(
      out_ent, filled, M_t, seq, toks, out_seq, out_pick, K, NP, T);
}